// ShiftedCorrelationLoss_1151051236036
// MI455X (gfx1250) — compile-verified
//
#include <hip/hip_runtime.h>

// Shifted correlation loss, restructured as a Toeplitz cross-correlation on the
// fp32 WMMA unit (V_WMMA_F32_16X16X4_F32), exact fp32 precision throughout.
//
// cov[b,s] = sum_t preds[b,t] * ym[b, t+128+s]  (no modulo wrap for |s|<=128)
// x_norm is shift-invariant; s = M + 16N - 128 maps 256 lags onto one 16x16
// WMMA tile via a Toeplitz operand decomposition; lag s=+128 is a scalar dot.

#define T_LEN   4096
#define L_PAD   4352          // T + 2*128
#define P_PHYS  4608          // logical [-128, 4480) with guard zeros
#define STEPS_PER_WAVE 136    // 8 waves x 136 = 1088 steps (>= 1084 needed);
                              // extra steps hit Pl's zero guard -> contribute 0

typedef __attribute__((ext_vector_type(2))) float v2f;
typedef __attribute__((ext_vector_type(8))) float v8f;

__launch_bounds__(256)
__global__ void shifted_corr_kernel(const float* __restrict__ preds,
                                    const float* __restrict__ labels,
                                    float* __restrict__ loss_per_b) {
    __shared__ float Yl[L_PAD + 16]; // ym; +16 tail pad for the guard steps
    __shared__ float Pl[P_PHYS];     // padded preds, physical = logical + 128
    __shared__ float red[2048];      // scalar reductions + 8-wave C-tile reduce

    const int b   = blockIdx.x;
    const int tid = threadIdx.x;
    const float* __restrict__ p = preds  + (size_t)b * T_LEN;
    const float* __restrict__ l = labels + (size_t)b * T_LEN;

    // ---- Phase 1: load rows into LDS, accumulate sums for means/norms ----
    float sp = 0.f, qp = 0.f, sl = 0.f, ql = 0.f;
    for (int t = tid; t < T_LEN; t += 256) {
        float pv = p[t];
        float lv = l[t];
        sp += pv; qp += pv * pv;
        sl += lv; ql += lv * lv;
        Pl[256 + t] = pv;          // logical index 128+t
        Yl[128 + t] = lv;
    }
    // zero guards: P physical [0,256) and [4352,4608); Y [0,128) and [4224,4352+16)
    Pl[tid]                = 0.f;
    Pl[P_PHYS - 256 + tid] = 0.f;
    if (tid < 128) {
        Yl[tid]               = 0.f;
        Yl[L_PAD - 128 + tid] = 0.f;
    }
    if (tid < 16) Yl[L_PAD + tid] = 0.f;

    // ---- Block-reduce the four sums ----
    red[tid]       = sp;
    red[256 + tid] = qp;
    red[512 + tid] = sl;
    red[768 + tid] = ql;
    __syncthreads();
    for (int off = 128; off > 0; off >>= 1) {
        if (tid < off) {
            red[tid]       += red[tid + off];
            red[256 + tid] += red[256 + tid + off];
            red[512 + tid] += red[512 + tid + off];
            red[768 + tid] += red[768 + tid + off];
        }
        __syncthreads();
    }
    const float Sp = red[0], Qp = red[256], Sl = red[512], Ql = red[768];
    const float invL   = 1.0f / (float)L_PAD;
    const float mu_l   = Sl * invL;
    const float xnorm2 = Qp - Sp * Sp * invL;   // shift-invariant ||xm||^2
    const float ynorm2 = Ql - Sl * Sl * invL;   // ||ym||^2
    const float inv_nm = 1.0f / (sqrtf(xnorm2) * sqrtf(ynorm2));
    __syncthreads();

    // ---- Center Y (pads become -mu; harmless since P is zero there) ----
    for (int i = tid; i < L_PAD; i += 256) Yl[i] -= mu_l;
    __syncthreads();

    // ---- Lag s = +128 via plain dot product (partial per thread) ----
    float c128 = 0.f;
    for (int t = tid; t < T_LEN; t += 256) c128 += Pl[256 + t] * Yl[256 + t];

    // ---- WMMA sweep: C[M,N] accumulates cov[s = M + 16N - 128] ----
    const int lane = tid & 31;
    const int wave = tid >> 5;
    const int half = lane >> 4;        // lanes 16-31 carry K = 2,3
    const int li   = lane & 15;        // M for A-operand, N for B-operand
    const int aoff = 2 * half + li;            // Y[u + k + M], k = 2*half + vgpr
    const int boff = 2 * half + 256 - 16 * li; // P_phys[u + k + 256 - 16N]
    const int ubase = wave * (STEPS_PER_WAVE * 4);  // contiguous chunk per wave

    v8f c = {};
    #pragma unroll 4
    for (int m = 0; m < STEPS_PER_WAVE; ++m) {
        const int u = ubase + m * 4;
        v2f a, bb;
        a.x  = Yl[u + aoff];
        a.y  = Yl[u + aoff + 1];
        bb.x = Pl[u + boff];
        bb.y = Pl[u + boff + 1];
        // D = A(16x4 f32) * B(4x16 f32) + C(16x16 f32)
        c = __builtin_amdgcn_wmma_f32_16x16x4_f32(
                /*neg_a=*/false, a, /*neg_b=*/false, bb,
                /*c_mod=*/(short)0, c, /*reuse_a=*/false, /*reuse_b=*/false);
    }

    // ---- Cross-wave reduction of the 8 partial tiles ----
    __syncthreads();
    #pragma unroll
    for (int r = 0; r < 8; ++r) red[wave * 256 + lane * 8 + r] = c[r];
    __syncthreads();

    float cov = 0.f;
    #pragma unroll
    for (int w = 0; w < 8; ++w) cov += red[w * 256 + tid];
    // element identity: tid = lane*8 + vgpr; s = M + 16*N - 128 (value only
    // matters for the max, so the (M,N) identity is implicit)
    float my_corr = cov * inv_nm;

    // ---- Sum-reduce c128, then max-reduce corr over the 257 shifts ----
    __syncthreads();
    red[tid] = c128;
    __syncthreads();
    for (int off = 128; off > 0; off >>= 1) {
        if (tid < off) red[tid] += red[tid + off];
        __syncthreads();
    }
    const float corr128 = red[0] * inv_nm;
    __syncthreads();
    red[tid] = my_corr;
    __syncthreads();
    for (int off = 128; off > 0; off >>= 1) {
        if (tid < off) red[tid] = fmaxf(red[tid], red[tid + off]);
        __syncthreads();
    }
    if (tid == 0) {
        loss_per_b[b] = 1.0f - fmaxf(red[0], corr128);
    }
}

__launch_bounds__(128)
__global__ void finalize_kernel(const float* __restrict__ loss_per_b,
                                float* __restrict__ out) {
    __shared__ float r[128];
    const int tid = threadIdx.x;
    r[tid] = loss_per_b[tid];
    __syncthreads();
    for (int off = 64; off > 0; off >>= 1) {
        if (tid < off) r[tid] += r[tid + off];
        __syncthreads();
    }
    if (tid == 0) out[0] = r[0] * (1.0f / 128.0f);
}

extern "C" void kernel_launch(void* const* d_in, const int* in_sizes, int n_in,
                              void* d_out, int out_size, void* d_ws, size_t ws_size,
                              hipStream_t stream) {
    const float* preds  = (const float*)d_in[0];
    const float* labels = (const float*)d_in[1];
    float* loss_per_b   = (float*)d_ws;       // 128 floats of scratch
    float* out          = (float*)d_out;

    shifted_corr_kernel<<<128, 256, 0, stream>>>(preds, labels, loss_per_b);
    finalize_kernel<<<1, 128, 0, stream>>>(loss_per_b, out);
}